// GraphSAGE_Allocation_Predictor_82609400971333
// MI455X (gfx1250) — compile-verified
//
#include <hip/hip_runtime.h>
#include <hip/hip_bf16.h>
#include <math.h>

#define N_NODES  100000
#define N_EDGES  3200000
#define N_GRAPHS 64
#define F_IN     9
#define H        64
#define HM       32   // H/2

typedef float v2f __attribute__((ext_vector_type(2)));
typedef float v8f __attribute__((ext_vector_type(8)));

// ---------------------------------------------------------------------------
// 0) zero the atomic accumulators (cnt | msg1 | msg2 | texp are contiguous)
// ---------------------------------------------------------------------------
__global__ void k_zero(float* __restrict__ p, int n) {
    int i = blockIdx.x * blockDim.x + threadIdx.x;
    if (i < n) p[i] = 0.0f;
}

// ---------------------------------------------------------------------------
// 1) layer-1 edge aggregation: cnt[d] += 1 ; msg1[d][:] += x[s][:]
// ---------------------------------------------------------------------------
__global__ void k_edge1(const int* __restrict__ src, const int* __restrict__ dst,
                        const float* __restrict__ x,
                        float* __restrict__ msg1, float* __restrict__ cnt) {
    int e = blockIdx.x * blockDim.x + threadIdx.x;
    if (e >= N_EDGES) return;
    int s = src[e], d = dst[e];
    atomicAdd(&cnt[d], 1.0f);
#pragma unroll
    for (int f = 0; f < F_IN; ++f)
        atomicAdd(&msg1[d * F_IN + f], x[s * F_IN + f]);
}

// ---------------------------------------------------------------------------
// 2) layer-1 linear (K = F_IN = 9, too small for WMMA): h1 = relu(lin)
// ---------------------------------------------------------------------------
__global__ void k_lin1(const float* __restrict__ msg1, const float* __restrict__ cnt,
                       const float* __restrict__ x,
                       const float* __restrict__ W1l, const float* __restrict__ b1l,
                       const float* __restrict__ W1r, float* __restrict__ h1) {
    int t = blockIdx.x * blockDim.x + threadIdx.x;
    if (t >= N_NODES * H) return;
    int n = t >> 6, o = t & 63;
    float rin = 1.0f / fmaxf(cnt[n], 1.0f);
    float s = b1l[o];
#pragma unroll
    for (int k = 0; k < F_IN; ++k)
        s += (msg1[n * F_IN + k] * rin) * W1l[o * F_IN + k]
           + x[n * F_IN + k] * W1r[o * F_IN + k];
    h1[t] = fmaxf(s, 0.0f);
}

// ---------------------------------------------------------------------------
// 3) layer-2 edge aggregation: one WAVE per edge.
//    readfirstlane makes the edge index wave-uniform -> src/dst become scalar
//    s_load; each lane moves a feature PAIR: one coalesced b64 load (the full
//    256B row per wave) + two f32 L2 atomics.
// ---------------------------------------------------------------------------
__global__ void k_edge2(const int* __restrict__ src, const int* __restrict__ dst,
                        const float* __restrict__ h1, float* __restrict__ msg2) {
    const int lane = threadIdx.x & 31;
    const int e = __builtin_amdgcn_readfirstlane(blockIdx.x * 8 + (threadIdx.x >> 5));
    if (e >= N_EDGES) return;
    const int s = src[e];
    const int d = dst[e];
    const v2f hv = *(const v2f*)(h1 + s * H + lane * 2);
    float* p = msg2 + d * H + lane * 2;
    atomicAdd(p, hv.x);
    atomicAdd(p + 1, hv.y);
}

// ---------------------------------------------------------------------------
// 4) layer-2 linear via V_WMMA_F32_16X16X4_F32 (full f32 precision):
//    h2[16n x 64] = agg2 @ W2l.T + h1 @ W2r.T + b2l
//    Weights staged once per workgroup into LDS in a K-paired layout so every
//    B fragment is a single aligned ds_load_b64.
//    LDS float index for Wt[k][col]:  (k>>1)*128 + col*2 + (k&1)
// ---------------------------------------------------------------------------
__global__ void __launch_bounds__(128) k_lin2(
        const float* __restrict__ msg2, const float* __restrict__ cnt,
        const float* __restrict__ h1,
        const float* __restrict__ W2l, const float* __restrict__ b2l,
        const float* __restrict__ W2r, float* __restrict__ h2) {
    __shared__ float lWl[H * H];
    __shared__ float lWr[H * H];

    // cooperative stage (all 128 threads, before any wave can exit)
    for (int idx = threadIdx.x; idx < H * H; idx += 128) {
        const int k   = idx & 63;
        const int col = idx >> 6;
        const int a   = (k >> 1) * 128 + col * 2 + (k & 1);
        lWl[a] = W2l[col * H + k];
        lWr[a] = W2r[col * H + k];
    }
    __syncthreads();

    const int lane = threadIdx.x & 31;
    const int wave = threadIdx.x >> 5;
    const int tile = blockIdx.x * 4 + wave;
    if (tile * 16 >= N_NODES) return;            // wave-uniform: EXEC stays full
    const int row  = lane & 15;
    const int hi   = lane >> 4;                  // lanes 16..31 hold K+2,K+3
    const int node = tile * 16 + row;
    const float rin = 1.0f / fmaxf(cnt[node], 1.0f);

    const float* m2r = msg2 + node * H;
    const float* h1r = h1   + node * H;
    const v2f* bWl = (const v2f*)lWl;            // v2f index: (k>>1)*64 + col
    const v2f* bWr = (const v2f*)lWr;

    v8f acc[4] = {};
#pragma unroll 4
    for (int k0 = 0; k0 < H; k0 += 4) {
        const int koff = k0 + 2 * hi;            // always even
        const int brow = (koff >> 1) * 64;
        v2f a1, a2;
        a1.x = m2r[koff]     * rin;              // A = agg2 tile (16x4)
        a1.y = m2r[koff + 1] * rin;
        a2.x = h1r[koff];                        // A' = h1 tile  (16x4)
        a2.y = h1r[koff + 1];
#pragma unroll
        for (int t = 0; t < 4; ++t) {
            const v2f b = bWl[brow + t * 16 + row];
            acc[t] = __builtin_amdgcn_wmma_f32_16x16x4_f32(
                         false, a1, false, b, (short)0, acc[t], false, false);
        }
#pragma unroll
        for (int t = 0; t < 4; ++t) {
            const v2f b = bWr[brow + t * 16 + row];
            acc[t] = __builtin_amdgcn_wmma_f32_16x16x4_f32(
                         false, a2, false, b, (short)0, acc[t], false, false);
        }
    }
    // C/D layout: VGPR r -> (M=r, N=lane) lanes0-15 ; (M=r+8, N=lane-16)
#pragma unroll
    for (int t = 0; t < 4; ++t) {
        const int col  = t * 16 + row;
        const float bv = b2l[col];
#pragma unroll
        for (int r = 0; r < 8; ++r) {
            const int nr = tile * 16 + r + 8 * hi;
            h2[nr * H + col] = acc[t][r] + bv;
        }
    }
}

// ---------------------------------------------------------------------------
// 5) MLP layer 1 via WMMA: m[16n x 32] = relu(h2 @ Wm1.T + bm1)
//    Wm1 (32x64) staged into LDS in the same K-paired layout.
// ---------------------------------------------------------------------------
__global__ void __launch_bounds__(128) k_mlp1(
        const float* __restrict__ h2,
        const float* __restrict__ Wm1, const float* __restrict__ bm1,
        float* __restrict__ m) {
    __shared__ float lW[HM * H];

    for (int idx = threadIdx.x; idx < HM * H; idx += 128) {
        const int k   = idx & 63;
        const int col = idx >> 6;                // 0..31
        lW[(k >> 1) * 64 + col * 2 + (k & 1)] = Wm1[col * H + k];
    }
    __syncthreads();

    const int lane = threadIdx.x & 31;
    const int wave = threadIdx.x >> 5;
    const int tile = blockIdx.x * 4 + wave;
    if (tile * 16 >= N_NODES) return;
    const int row  = lane & 15;
    const int hi   = lane >> 4;
    const int node = tile * 16 + row;
    const float* h2r = h2 + node * H;
    const v2f* bW = (const v2f*)lW;              // v2f index: (k>>1)*32 + col

    v8f acc[2] = {};
#pragma unroll 4
    for (int k0 = 0; k0 < H; k0 += 4) {
        const int koff = k0 + 2 * hi;            // even
        const int brow = (koff >> 1) * 32;
        v2f a;
        a.x = h2r[koff];
        a.y = h2r[koff + 1];
#pragma unroll
        for (int t = 0; t < 2; ++t) {
            const v2f b = bW[brow + t * 16 + row];
            acc[t] = __builtin_amdgcn_wmma_f32_16x16x4_f32(
                         false, a, false, b, (short)0, acc[t], false, false);
        }
    }
#pragma unroll
    for (int t = 0; t < 2; ++t) {
        const int col  = t * 16 + row;
        const float bv = bm1[col];
#pragma unroll
        for (int r = 0; r < 8; ++r) {
            const int nr = tile * 16 + r + 8 * hi;
            m[nr * HM + col] = fmaxf(acc[t][r] + bv, 0.0f);
        }
    }
}

// ---------------------------------------------------------------------------
// 6) pi_raw = sigmoid(m @ Wm2.T + bm2) ; texp[batch] += pi_raw
// ---------------------------------------------------------------------------
__global__ void k_pi(const float* __restrict__ m, const float* __restrict__ Wm2,
                     const float* __restrict__ bm2, const int* __restrict__ batch,
                     float* __restrict__ pi, float* __restrict__ texp) {
    int n = blockIdx.x * blockDim.x + threadIdx.x;
    if (n >= N_NODES) return;
    float s = bm2[0];
#pragma unroll
    for (int k = 0; k < HM; ++k) s += m[n * HM + k] * Wm2[k];
    float p = 1.0f / (1.0f + expf(-s));
    pi[n] = p;
    atomicAdd(&texp[batch[n]], p);
}

// ---------------------------------------------------------------------------
// 7) out = pi_raw * min(B_total[g] / (texp[g] + 1e-12), 1)
// ---------------------------------------------------------------------------
__global__ void k_final(const float* __restrict__ pi, const int* __restrict__ batch,
                        const float* __restrict__ B_total, const float* __restrict__ texp,
                        float* __restrict__ out) {
    int n = blockIdx.x * blockDim.x + threadIdx.x;
    if (n >= N_NODES) return;
    int g = batch[n];
    float ratio = fminf(B_total[g] / (texp[g] + 1e-12f), 1.0f);
    out[n] = pi[n] * ratio;
}

// ---------------------------------------------------------------------------
extern "C" void kernel_launch(void* const* d_in, const int* in_sizes, int n_in,
                              void* d_out, int out_size, void* d_ws, size_t ws_size,
                              hipStream_t stream) {
    (void)in_sizes; (void)n_in; (void)out_size; (void)ws_size;

    const float* x       = (const float*)d_in[0];
    const int*   ei      = (const int*)  d_in[1];
    const int*   src     = ei;
    const int*   dst     = ei + N_EDGES;
    // d_in[2] = edge_attr (unused by reference)
    const int*   batch   = (const int*)  d_in[3];
    const float* B_total = (const float*)d_in[4];
    const float* W1l     = (const float*)d_in[5];
    const float* b1l     = (const float*)d_in[6];
    const float* W1r     = (const float*)d_in[7];
    const float* W2l     = (const float*)d_in[8];
    const float* b2l     = (const float*)d_in[9];
    const float* W2r     = (const float*)d_in[10];
    const float* Wm1     = (const float*)d_in[11];
    const float* bm1     = (const float*)d_in[12];
    const float* Wm2     = (const float*)d_in[13];
    const float* bm2     = (const float*)d_in[14];
    float*       out     = (float*)d_out;

    // workspace layout (floats); atomic accumulators first (contiguous zeroing)
    float* ws   = (float*)d_ws;
    float* cnt  = ws;                                   // N
    float* msg1 = cnt  + N_NODES;                       // 9N
    float* msg2 = msg1 + (size_t)N_NODES * F_IN;        // 64N
    float* texp = msg2 + (size_t)N_NODES * H;           // 64
    float* h1   = texp + N_GRAPHS;                      // 64N
    float* h2   = h1   + (size_t)N_NODES * H;           // 64N
    float* mbuf = h2   + (size_t)N_NODES * H;           // 32N
    float* pi   = mbuf + (size_t)N_NODES * HM;          // N

    const int nzero = N_NODES * (1 + F_IN + H) + N_GRAPHS;
    k_zero<<<(nzero + 255) / 256, 256, 0, stream>>>(cnt, nzero);

    k_edge1<<<(N_EDGES + 255) / 256, 256, 0, stream>>>(src, dst, x, msg1, cnt);

    k_lin1<<<(N_NODES * H + 255) / 256, 256, 0, stream>>>(msg1, cnt, x, W1l, b1l, W1r, h1);

    // one wave per edge, 8 waves per 256-thread block
    k_edge2<<<N_EDGES / 8, 256, 0, stream>>>(src, dst, h1, msg2);

    const int ntiles = (N_NODES + 15) / 16;             // 6250
    k_lin2<<<(ntiles + 3) / 4, 128, 0, stream>>>(msg2, cnt, h1, W2l, b2l, W2r, h2);

    k_mlp1<<<(ntiles + 3) / 4, 128, 0, stream>>>(h2, Wm1, bm1, mbuf);

    k_pi<<<(N_NODES + 255) / 256, 256, 0, stream>>>(mbuf, Wm2, bm2, batch, pi, texp);

    k_final<<<(N_NODES + 255) / 256, 256, 0, stream>>>(pi, batch, B_total, texp, out);
}